// loss_19585050870269
// MI455X (gfx1250) — compile-verified
//
#include <hip/hip_runtime.h>
#include <math.h>

#define BB 64
#define AA 8732
#define CC 81
#define DD 85            // 4 loc + 81 cls
#define EPSV 1e-7f
#define TILE_M 16
#define WPB 2            // waves per block in pass1 (546 tiles = 273 blocks exactly)
#define TILE_E (TILE_M * DD)
#define TILES ((AA + TILE_M - 1) / TILE_M)   // 546
#define NSORT 16384

typedef float v2f __attribute__((ext_vector_type(2)));
typedef float v8f __attribute__((ext_vector_type(8)));

// ---------------- init: zero the 4*B accumulators (npos, fg, loc, bg) -------
__global__ void zero_acc_k(float* acc) {
    int i = threadIdx.x;
    if (i < 4 * BB) acc[i] = 0.f;
}

// ---------------- pass 1: stream pred/gt, entropy via WMMA softmax-sum ------
__global__ __launch_bounds__(WPB * 32) void pass1_k(
    const float* __restrict__ pred, const float* __restrict__ gt,
    float* __restrict__ eneg, float* __restrict__ acc)
{
    __shared__ float sp[WPB][TILE_E];
    __shared__ float sg[WPB][TILE_E];

    const int wave = threadIdx.x >> 5;
    const int lane = threadIdx.x & 31;
    const int bpb  = TILES / WPB;            // 273 blocks per batch
    const int b    = blockIdx.x / bpb;
    const int grp  = blockIdx.x % bpb;
    const int tile = grp * WPB + wave;       // always < 546
    const int a0   = tile * TILE_M;

    const long base  = (long)b * (long)(AA * DD) + (long)a0 * DD;
    const int  valid = (AA - a0) * DD;       // clamp for the ragged last tile
    float* lp = sp[wave];
    float* lg = sg[wave];

    // Coalesced stage of the contiguous 16x85 tile (pred & gt) into LDS.
    for (int i = lane; i < TILE_E; i += 32) {
        float vp = 0.f, vg = 0.f;
        if (i < valid) { vp = pred[base + i]; vg = gt[base + i]; }
        lp[i] = vp; lg[i] = vg;
    }
    // Same-wave DS ordering: compiler inserts the s_wait_dscnt for its loads.

    const int  row = lane & 15;
    const int  half = lane >> 4;             // matches f32 WMMA A-layout halves
    const int  a = a0 + row;
    const bool rowvalid = (a < AA);
    const float* prow = lp + row * DD;
    const float* grow = lg + row * DD;

    // Split the 81 class columns (4..84) across the half-waves: max + label dot.
    int c0 = half ? 45 : 4;
    int c1 = half ? 85 : 45;
    float m = -1e30f, dsel = 0.f;
    for (int c = c0; c < c1; ++c) {
        float x = prow[c];
        m = fmaxf(m, x);
        dsel += grow[c] * x;                 // one-hot dot -> logit at label
    }
    m    = fmaxf(m, __shfl_xor(m, 16, 32));
    dsel += __shfl_xor(dsel, 16, 32);
    float neg = grow[4];                     // cls_gt[..,0]

    // WMMA fp32 row-sum of exp(x-m): A = 16x4 chunk, B = ones, D += A*B.
    // f32 A layout: lanes 0-15 hold (K0,K1), lanes 16-31 hold (K2,K3) of the
    // same 16 rows -> k = 4*chunk + 2*half, components {k, k+1}.
    v8f cacc = {0.f, 0.f, 0.f, 0.f, 0.f, 0.f, 0.f, 0.f};
    v2f bones; bones.x = 1.f; bones.y = 1.f;
    for (int ch = 0; ch < 21; ++ch) {        // K = 84 (81 real + 3 zero pad)
        int k = ch * 4 + half * 2;
        float a0v = (k     < CC) ? __expf(prow[4 + k]     - m) : 0.f;
        float a1v = (k + 1 < CC) ? __expf(prow[4 + k + 1] - m) : 0.f;
        v2f av; av.x = a0v; av.y = a1v;
        cacc = __builtin_amdgcn_wmma_f32_16x16x4_f32(
            /*neg_a=*/false, av, /*neg_b=*/false, bones,
            /*c_mod=*/(short)0, cacc, /*reuse_a=*/false, /*reuse_b=*/false);
    }
    // D layout: VGPR v on lanes 0-15 = row v, on lanes 16-31 = row v+8; all
    // columns equal (B = ones). Lane l's component (l&7) is the row sum of
    // row (l&7)+8*(l>=16); fetch our own row's sum with one shuffle.
    float comp[8];
#pragma unroll
    for (int i = 0; i < 8; ++i) comp[i] = cacc[i];
    float scand = comp[lane & 7];
    int holder  = (row < 8) ? row : (row + 8);
    float s = __shfl(scand, holder, 32);

    // entropy at the label, clipped like the reference
    float p = __expf(dsel - m) / s;
    p = fminf(fmaxf(p, EPSV), 1.f - EPSV);
    float e   = -__logf(p);
    float pos = 1.f - neg;
    float hub = 0.25f * (fabsf(prow[0] - grow[0]) + fabsf(prow[1] - grow[1]) +
                         fabsf(prow[2] - grow[2]) + fabsf(prow[3] - grow[3]));
    float en = e * neg;
    float ep = e * pos;
    float pl = pos * hub;

    if (half == 0 && rowvalid) {
        eneg[(long)b * AA + a] = en;
    } else {                                  // avoid double counting half 1
        ep = 0.f; pl = 0.f; pos = 0.f;
    }
    if (!rowvalid) { ep = 0.f; pl = 0.f; pos = 0.f; }

    // wave reduction, then 3 atomics per wave
    for (int off = 16; off > 0; off >>= 1) {
        pos += __shfl_xor(pos, off, 32);
        ep  += __shfl_xor(ep,  off, 32);
        pl  += __shfl_xor(pl,  off, 32);
    }
    if (lane == 0) {
        atomicAdd(&acc[b],          pos);     // npos
        atomicAdd(&acc[BB + b],     ep);      // loss_fg
        atomicAdd(&acc[2 * BB + b], pl);      // loss_loc
    }
}

// ---------------- pass 2: per-batch top-(3*npos) sum via LDS bitonic sort ---
__global__ void pass2_k(const float* __restrict__ eneg,
                        const float* __restrict__ acc,
                        float* __restrict__ bg)
{
    extern __shared__ float sm[];            // NSORT floats = 64 KB dynamic LDS
    const int b   = blockIdx.x;
    const int tid = threadIdx.x;
    const int nt  = blockDim.x;

    for (int i = tid; i < NSORT; i += nt)
        sm[i] = (i < AA) ? eneg[(long)b * AA + i] : -1.f;   // e_neg >= 0
    __syncthreads();

    // bitonic sort, descending
    for (int k = 2; k <= NSORT; k <<= 1) {
        for (int j = k >> 1; j > 0; j >>= 1) {
            for (int i = tid; i < NSORT; i += nt) {
                int ixj = i ^ j;
                if (ixj > i) {
                    float x = sm[i], y = sm[ixj];
                    bool descSeg = ((i & k) == 0);
                    if (descSeg ? (x < y) : (x > y)) { sm[i] = y; sm[ixj] = x; }
                }
            }
            __syncthreads();
        }
    }

    float thres = 3.f * acc[b];              // npos * 3
    float part = 0.f;
    for (int i = tid; i < AA; i += nt)
        if ((float)i < thres) part += sm[i];
    __syncthreads();
    sm[tid] = part;
    __syncthreads();
    for (int s = nt >> 1; s > 0; s >>= 1) {
        if (tid < s) sm[tid] += sm[tid + s];
        __syncthreads();
    }
    if (tid == 0) bg[b] = sm[0];
}

// ---------------- pass 3: the three scalar means ----------------------------
__global__ void pass3_k(const float* __restrict__ acc,
                        const float* __restrict__ bg,
                        float* __restrict__ out)
{
    __shared__ float ra[BB], rc[BB], rl[BB];
    int b = threadIdx.x;                     // blockDim.x == BB == 64
    float np = acc[b], fg = acc[BB + b], ll = acc[2 * BB + b], lb = bg[b];
    float lc = fg + lb;
    ra[b] = (lc + ll) / np;
    rc[b] = lc / np;
    rl[b] = ll / np;
    __syncthreads();
    for (int s = BB >> 1; s > 0; s >>= 1) {
        if (b < s) { ra[b] += ra[b + s]; rc[b] += rc[b + s]; rl[b] += rl[b + s]; }
        __syncthreads();
    }
    if (b == 0) {
        out[0] = ra[0] / (float)BB;   // loss_all
        out[1] = rc[0] / (float)BB;   // loss_cls_m
        out[2] = rl[0] / (float)BB;   // loss_loc_m
    }
}

extern "C" void kernel_launch(void* const* d_in, const int* in_sizes, int n_in,
                              void* d_out, int out_size, void* d_ws, size_t ws_size,
                              hipStream_t stream)
{
    const float* pred = (const float*)d_in[0];
    const float* gt   = (const float*)d_in[1];

    float* ws   = (float*)d_ws;
    float* eneg = ws;                          // B*A floats (~2.23 MB)
    float* acc  = ws + (size_t)BB * AA;        // 3*B: npos, loss_fg, loss_loc
    float* bg   = acc + 3 * BB;                // B:   loss_bg
    float* out  = (float*)d_out;

    zero_acc_k<<<1, 256, 0, stream>>>(acc);    // zeroes acc+bg (4*B = 256)

    const int bpb = TILES / WPB;               // 273
    pass1_k<<<BB * bpb, WPB * 32, 0, stream>>>(pred, gt, eneg, acc);

    pass2_k<<<BB, 256, NSORT * sizeof(float), stream>>>(eneg, acc, bg);

    pass3_k<<<1, BB, 0, stream>>>(acc, bg, out);
}